// Convolution_5755256176641
// MI455X (gfx1250) — compile-verified
//
#include <hip/hip_runtime.h>
#include <cstdint>

#define HH 4096
#define WW 4096
#define TILE 64
#define PAD 8
#define RAW 80          // TILE + 2*PAD
#define SATP 81         // odd pitch -> conflict-free column access (gcd(81,64)=1)
#define NTHREADS 256

// ---- CDNA5 async global->LDS copy path (tracked by ASYNCcnt) ----
__device__ __forceinline__ void async_load_b32(uint32_t lds_off, uint64_t gaddr) {
    asm volatile("global_load_async_to_lds_b32 %0, %1, off"
                 :: "v"(lds_off), "v"(gaddr) : "memory");
}
__device__ __forceinline__ void async_load_b128(uint32_t lds_off, uint64_t gaddr) {
    asm volatile("global_load_async_to_lds_b128 %0, %1, off"
                 :: "v"(lds_off), "v"(gaddr) : "memory");
}
__device__ __forceinline__ void wait_async0() {
    asm volatile("s_wait_asynccnt 0" ::: "memory");
}

__global__ __launch_bounds__(NTHREADS)
void multibox_sat_kernel(const float* __restrict__ x,
                         const float* __restrict__ base,
                         float* __restrict__ out) {
    __shared__ __align__(16) float raw[RAW * RAW];
    __shared__ __align__(16) float sat[SATP * SATP];

    const int tid = threadIdx.x;
    const int bx  = blockIdx.x;
    const int by  = blockIdx.y;
    const int gx0 = bx * TILE - PAD;
    const int gy0 = by * TILE - PAD;

    const uint32_t raw_base = (uint32_t)(uintptr_t)raw;  // low 32 bits = LDS byte offset

    const bool interior = (gx0 >= 0) && (gy0 >= 0) &&
                          (gx0 + RAW <= WW) && (gy0 + RAW <= HH);

    if (interior) {
        // 80 rows x 20 float4 groups; global 16B-aligned (gx0*4 = bx*256-32),
        // LDS dest 16B-aligned (pitch 80 floats = 320 B).
        const uint64_t gbase = (uint64_t)(uintptr_t)(x + (size_t)gy0 * WW + gx0);
        for (int idx = tid; idx < RAW * 20; idx += NTHREADS) {
            const int r  = idx / 20;
            const int cg = idx % 20;
            const uint32_t lds = raw_base + (uint32_t)((r * RAW + cg * 4) * 4);
            const uint64_t ga  = gbase + ((uint64_t)((size_t)r * WW + (size_t)cg * 4) << 2);
            async_load_b128(lds, ga);
        }
    } else {
        // Border tiles: clamped per-element loads == 'edge' padding semantics.
        for (int idx = tid; idx < RAW * RAW; idx += NTHREADS) {
            const int r = idx / RAW;
            const int c = idx % RAW;
            const int gy = min(max(gy0 + r, 0), HH - 1);
            const int gx = min(max(gx0 + c, 0), WW - 1);
            const uint32_t lds = raw_base + (uint32_t)(idx * 4);
            const uint64_t ga  = (uint64_t)(uintptr_t)(x + (size_t)gy * WW + gx);
            async_load_b32(lds, ga);
        }
    }
    wait_async0();
    __syncthreads();

    // Phase 1: vertical (column) running sums -> sat[r+1][c+1] = sum_{r'<=r} raw[r'][c]
    // Lanes read consecutive columns (stride 1): conflict-free.
    if (tid < RAW) {
        const int c = tid;
        float s = 0.f;
        #pragma unroll 4
        for (int r = 0; r < RAW; ++r) {
            s += raw[r * RAW + c];
            sat[(r + 1) * SATP + (c + 1)] = s;
        }
    }
    __syncthreads();

    // Phase 2: horizontal prefix in place -> full SAT.
    // Lanes read column-direction with stride 81 (gcd(81,64)=1): conflict-free.
    if (tid < RAW) {
        const int r = tid + 1;
        float s = 0.f;
        #pragma unroll 4
        for (int c = 1; c <= RAW; ++c) {
            s += sat[r * SATP + c];
            sat[r * SATP + c] = s;
        }
    }
    __syncthreads();

    // Phase 3: each pixel = sum_k (1/(7 k^2)) * SAT 4-corner box.
    const float wgt[7] = {
        1.f / (9.f   * 7.f), 1.f / (25.f  * 7.f), 1.f / (49.f  * 7.f),
        1.f / (81.f  * 7.f), 1.f / (121.f * 7.f), 1.f / (169.f * 7.f),
        1.f / (225.f * 7.f)
    };

    #pragma unroll 4
    for (int i = 0; i < (TILE * TILE) / NTHREADS; ++i) {
        const int id = i * NTHREADS + tid;
        const int ty = id / TILE;
        const int tx = id % TILE;          // lanes -> consecutive tx: stride-1 LDS + coalesced global

        float acc = 0.f;
        #pragma unroll
        for (int p = 1; p <= 7; ++p) {
            const int r1 = (ty + 9 + p) * SATP;   // rows [ty+8-p, ty+8+p]
            const int r0 = (ty + 8 - p) * SATP;
            const int c1 = tx + 9 + p;            // cols [tx+8-p, tx+8+p]
            const int c0 = tx + 8 - p;
            const float s = sat[r1 + c1] - sat[r1 + c0]
                          - sat[r0 + c1] + sat[r0 + c0];
            acc += wgt[p - 1] * s;
        }

        const int gy = by * TILE + ty;
        const int gx = bx * TILE + tx;
        const size_t o = (size_t)gy * WW + gx;
        // base/out are touched exactly once: non-temporal so they don't evict
        // the x-tile halos from L2 (halo rows are reused by adjacent blocks).
        const float b = __builtin_nontemporal_load(base + o);
        __builtin_nontemporal_store(acc * b, out + o);
    }
}

extern "C" void kernel_launch(void* const* d_in, const int* in_sizes, int n_in,
                              void* d_out, int out_size, void* d_ws, size_t ws_size,
                              hipStream_t stream) {
    (void)in_sizes; (void)n_in; (void)out_size; (void)d_ws; (void)ws_size;
    const float* x    = (const float*)d_in[0];
    const float* base = (const float*)d_in[1];
    float* out        = (float*)d_out;

    dim3 grid(WW / TILE, HH / TILE);   // 64 x 64 tiles
    multibox_sat_kernel<<<grid, NTHREADS, 0, stream>>>(x, base, out);
}